// DoubleIntegralHatNew_39548058862251
// MI455X (gfx1250) — compile-verified
//
#include <hip/hip_runtime.h>
#include <math.h>
#include <stdint.h>

// DoubleIntegralHatNew for MI455X (gfx1250).
// Compute-bound (>20x over HBM at 23.3 TB/s: 16.8 MB moved vs ~1.7G VALU ops
// + ~200M transcendentals). Strategy: fp32 VALU with hardware exp2/rcp
// sigmoids (log2e folded into the output-layer weights), per-channel params
// and hoisted inner-layer bias points in registers, x tile staged via CDNA5
// async global->LDS DMA overlapped with the per-channel fi1/fi2 preamble.

#define TPB    128   // 4 wave32 per block (128 channels per block)
#define ROWS   32    // batch rows per block -> preamble overhead 2/32
#define NB     20
#define NNODES 21
#define HID    5

#define LOG2E  1.44269504088896340736f
#define KTEMP  (50.0f * LOG2E)          // mask temperature in exp2 domain

struct Quad { float t[NNODES]; float u[NNODES]; float w[NNODES]; };

__device__ __forceinline__ float fast_sigmoid(float s) {
  // sigmoid(s) = 1 / (1 + 2^(-s*log2e))  -> v_exp_f32 + v_rcp_f32
  float e = __builtin_amdgcn_exp2f(-LOG2E * s);
  return __builtin_amdgcn_rcpf(1.0f + e);
}

// Quadrature core: acc = sum_n ww[n] * sigmoid-from-exp2( p + q*tt[n] ).
// wos/bos are the output-layer weights pre-scaled by -log2e, so the per-node
// sigmoid is just rcp(1 + exp2(s)) with zero extra multiplies.
__device__ __forceinline__ float cheb_core(
    const float p[HID], const float qv[HID],
    const float wos[HID], float bos,
    const float* __restrict__ tt, const float* __restrict__ ww) {
  float acc = 0.0f;
#pragma unroll
  for (int n = 0; n < NNODES; ++n) {
    float s = bos;
#pragma unroll
    for (int h = 0; h < HID; ++h) {
      float v = fmaf(qv[h], tt[n], p[h]);
      s = fmaf(wos[h], fmaxf(v, 0.0f), s);
    }
    float e = __builtin_amdgcn_exp2f(s);
    acc = fmaf(ww[n], __builtin_amdgcn_rcpf(1.0f + e), acc);
  }
  return acc;
}

// General (non-hoisted) integral over [x0, x0+dx] for the per-channel preamble.
__device__ __forceinline__ float cheb_integral(
    float x0, float dx,
    const float wi[HID], const float bi[HID], const float wos[HID], float bos,
    const Quad& q) {
  float p[HID], r[HID];
#pragma unroll
  for (int h = 0; h < HID; ++h) {
    p[h] = fmaf(wi[h], x0, bi[h]);
    r[h] = wi[h] * dx;
  }
  return cheb_core(p, r, wos, bos, q.t, q.w) * dx * 0.5f;
}

__global__ __launch_bounds__(TPB) void hat_kernel(
    const float* __restrict__ X,  const float* __restrict__ A,
    const float* __restrict__ Bv, const float* __restrict__ M,
    const float* __restrict__ S,
    const float* __restrict__ w1i, const float* __restrict__ b1i,
    const float* __restrict__ w1o, const float* __restrict__ b1o,
    const float* __restrict__ w2i, const float* __restrict__ b2i,
    const float* __restrict__ w2o, const float* __restrict__ b2o,
    float* __restrict__ OUT, int Bn, int Cn, Quad q) {
  __shared__ float xs[ROWS * TPB];

  const int tid = threadIdx.x;
  const int c   = blockIdx.x * TPB + tid;
  const int b0  = blockIdx.y * ROWS;
  if (c >= Cn) return;

  int nrows = Bn - b0;
  if (nrows > ROWS) nrows = ROWS;

  // ---- CDNA5 async DMA: stage this block's x tile (ROWS x TPB) into LDS.
  // Issued up front so the transfers overlap the fi1/fi2 preamble below.
  // Lane-private readback -> only an ASYNCcnt wait is needed (no barrier).
  const uint32_t lds_base = (uint32_t)(uintptr_t)xs + (uint32_t)(tid * 4u);
  for (int r = 0; r < nrows; ++r) {
    const float* gp = X + (size_t)(b0 + r) * (size_t)Cn + (size_t)c;
    uint32_t la = lds_base + (uint32_t)(r * TPB * 4);
    asm volatile("global_load_async_to_lds_b32 %0, %1, off"
                 :: "v"(la), "v"(gp) : "memory");
  }

  // ---- Per-channel parameters -> registers.
  float wi1[HID], bi1[HID], wos1[HID], wi2[HID], bi2[HID], wos2[HID];
#pragma unroll
  for (int h = 0; h < HID; ++h) {
    wi1[h] = w1i[c * HID + h]; bi1[h] = b1i[c * HID + h];
    wos1[h] = w1o[c * HID + h] * (-LOG2E);          // fold sigmoid scale
    wi2[h] = w2i[c * HID + h]; bi2[h] = b2i[c * HID + h];
    wos2[h] = w2o[c * HID + h] * (-LOG2E);
  }
  float bos1 = b1o[c] * (-LOG2E), bos2 = b2o[c] * (-LOG2E);

  float av    = A[c];
  float width = fabsf(Bv[c]);
  float left  = av;
  float mid   = fmaf(fast_sigmoid(M[c]), width, av);
  float right = av + width;
  float height = fabsf(S[c]);

  // Per-channel normalization integrals + safe_div (amortized over ROWS rows).
  float fi1 = cheb_integral(left, mid - left,  wi1, bi1, wos1, bos1, q);
  float fi2 = cheb_integral(mid,  right - mid, wi2, bi2, wos2, bos2, q);
  float sgn = (fi2 < 0.0f) ? -1.0f : 1.0f;   // sign(0) -> +1 per reference
  float sf  = fi1 / (fi2 + sgn * 1e-10f);
  sf = fminf(fmaxf(sf, -100000.0f), 100000.0f);

  // Hoisted inner-layer bias points: integral_left is anchored at `left`,
  // integral_right re-anchored at `right` (X_n = right + (lr-right)*u_n,
  // u_n = 1 - t_n), so both p-vectors are row-invariant.
  float p1[HID], p2[HID];
#pragma unroll
  for (int h = 0; h < HID; ++h) {
    p1[h] = fmaf(wi1[h], left,  bi1[h]);
    p2[h] = fmaf(wi2[h], right, bi2[h]);
  }
  // Mask arguments in exp2 domain, hoisted.
  float Kl = KTEMP * left, Km = KTEMP * mid, Kr = KTEMP * right;

  // Wait for the async x tile, then burn through the rows.
  asm volatile("s_wait_asynccnt 0x0" ::: "memory");

  for (int r = 0; r < nrows; ++r) {
    float x  = xs[r * TPB + tid];
    float ul = fmaxf(x, left);    // relu(x-left)+left
    float lr = fminf(x, right);   // right-relu(right-x)

    float dxl = ul - left;        // >= 0
    float dxr = lr - right;       // <= 0 (right-anchored)
    float q1[HID], q2[HID];
#pragma unroll
    for (int h = 0; h < HID; ++h) {
      q1[h] = wi1[h] * dxl;
      q2[h] = wi2[h] * dxr;
    }
    float il = cheb_core(p1, q1, wos1, bos1, q.t, q.w) * dxl * 0.5f;
    float ir = cheb_core(p2, q2, wos2, bos2, q.u, q.w) * (-dxr) * 0.5f;

    // sig(a)*sig(b) = rcp(1 + ea + eb + ea*eb), args prehoisted.
    float Kx = KTEMP * x;
    float ea = __builtin_amdgcn_exp2f(Kl - Kx);
    float eb = __builtin_amdgcn_exp2f(Kx - Km);
    float m1 = __builtin_amdgcn_rcpf(fmaf(ea, eb, ea + eb + 1.0f));
    float ec = __builtin_amdgcn_exp2f(Km - Kx);
    float ed = __builtin_amdgcn_exp2f(Kx - Kr);
    float m2 = __builtin_amdgcn_rcpf(fmaf(ec, ed, ec + ed + 1.0f));

    OUT[(size_t)(b0 + r) * (size_t)Cn + (size_t)c] =
        fmaf(il, m1, sf * ir * m2) * height;
  }
}

// Host-side replication of the numpy Clenshaw-Curtis construction (float64,
// cast to float32 exactly like the reference).
static Quad make_quad() {
  Quad q;
  double lam[NNODES][NNODES];
  for (int i = 0; i < NNODES; ++i)
    for (int j = 0; j < NNODES; ++j)
      lam[i][j] = cos((double)i * (double)j * M_PI / (double)NB);
  for (int i = 0; i < NNODES; ++i) {
    lam[i][0] = 0.5;
    lam[i][NB] *= 0.5;
  }
  for (int i = 0; i < NNODES; ++i)
    for (int j = 0; j < NNODES; ++j)
      lam[i][j] *= 2.0 / (double)NB;

  double W[NNODES];
  for (int i = 0; i < NNODES; ++i) {
    if (i == 0)      W[i] = 1.0;
    else if (i & 1)  W[i] = 0.0;
    else             W[i] = 2.0 / (1.0 - (double)i * (double)i);
  }
  for (int j = 0; j < NNODES; ++j) {   // cc_w = lam.T @ W
    double s = 0.0;
    for (int i = 0; i < NNODES; ++i) s += lam[i][j] * W[i];
    q.w[j] = (float)s;
  }
  for (int n = 0; n < NNODES; ++n) {   // t = (STEPS + 1) * 0.5, STEPS in f32
    float st = (float)cos((double)n * M_PI / (double)NB);
    q.t[n] = (st + 1.0f) * 0.5f;
    q.u[n] = 1.0f - q.t[n];
  }
  return q;
}

extern "C" void kernel_launch(void* const* d_in, const int* in_sizes, int n_in,
                              void* d_out, int out_size, void* d_ws, size_t ws_size,
                              hipStream_t stream) {
  (void)n_in; (void)out_size; (void)d_ws; (void)ws_size;
  const float* X   = (const float*)d_in[0];
  const float* A   = (const float*)d_in[1];
  const float* Bv  = (const float*)d_in[2];
  const float* M   = (const float*)d_in[3];
  const float* S   = (const float*)d_in[4];
  const float* w1i = (const float*)d_in[5];
  const float* b1i = (const float*)d_in[6];
  const float* w1o = (const float*)d_in[7];
  const float* b1o = (const float*)d_in[8];
  const float* w2i = (const float*)d_in[9];
  const float* b2i = (const float*)d_in[10];
  const float* w2o = (const float*)d_in[11];
  const float* b2o = (const float*)d_in[12];

  const int Cn = in_sizes[1];              // a has shape (C,)
  const int Bn = in_sizes[0] / Cn;         // x has shape (B, C)

  Quad q = make_quad();                    // deterministic host math

  dim3 grid((Cn + TPB - 1) / TPB, (Bn + ROWS - 1) / ROWS);
  hat_kernel<<<grid, TPB, 0, stream>>>(X, A, Bv, M, S,
                                       w1i, b1i, w1o, b1o,
                                       w2i, b2i, w2o, b2o,
                                       (float*)d_out, Bn, Cn, q);
}